// GCN_10256381903258
// MI455X (gfx1250) — compile-verified
//
#include <hip/hip_runtime.h>

typedef __attribute__((ext_vector_type(2))) float v2f;
typedef __attribute__((ext_vector_type(8))) float v8f;

#define N_NODES 100000
#define D_IN    128
#define D_HID   64
#define D_OUT   64
#define N_EDGES 1600000

// LDS row stride (in float2) for the staged W pairs. 80 float2 = 640 bytes
// per k-pair row => consecutive k-pair rows start 160 banks (mod 64 = 32
// banks) apart, so lanes 0-15 (even pair row) hit banks 0..31 while lanes
// 16-31 (odd pair row) hit banks 32..63: conflict-free ds_load_b64.
#define WP_STRIDE 80

// -------------------------------------------------------------------------
// Persistent wave-per-tile fp32 WMMA GEMM: out[n][64] = A[n][K] @ W[K][64]
// + bias. W is staged once per block into LDS as float2 pairs
// Wp[k/2][col] = {W[k][col], W[k+1][col]}, then each WMMA B-operand is one
// ds_load_b64. Waves grid-stride over 16-row tiles (wave-uniform bounds so
// EXEC is all-1s for every V_WMMA).
// -------------------------------------------------------------------------
template <int K>
__global__ __launch_bounds__(256) void gemm_wmma_f32(
    const float* __restrict__ A, const float* __restrict__ W,
    const float* __restrict__ bias, float* __restrict__ out, int nRows,
    int nBlocks) {
  __shared__ float2 Wp[(K / 2) * WP_STRIDE];

  // ---- stage W into LDS (coalesced global reads, conflict-free LDS writes)
  for (int i = threadIdx.x; i < K * 64; i += 256) {
    const int k = i >> 6;   // W row
    const int c = i & 63;   // W col
    ((float*)Wp)[(k >> 1) * (2 * WP_STRIDE) + c * 2 + (k & 1)] = W[i];
  }
  __syncthreads();

  const int lane = threadIdx.x & 31;
  const int wave = threadIdx.x >> 5;
  const int half = lane >> 4;  // 0: K pair {0,1}, 1: K pair {2,3}
  const int l15  = lane & 15;  // row (A) / col (B,C,D)

  const float b0 = bias[l15 +  0];
  const float b1 = bias[l15 + 16];
  const float b2 = bias[l15 + 32];
  const float b3 = bias[l15 + 48];

  const int nTiles = nRows >> 4;  // 16-row tiles (100000 % 16 == 0)

  for (int tile = blockIdx.x * 8 + wave; tile < nTiles;
       tile += nBlocks * 8) {  // wave-uniform stride
    const int rowBase = tile * 16;
    const float* aRow = A + (size_t)(rowBase + l15) * K;

    v8f acc0 = {}, acc1 = {}, acc2 = {}, acc3 = {};

#pragma unroll 4
    for (int k = 0; k < K; k += 4) {
      const int kk = k + 2 * half;
      v2f a;
      a.x = aRow[kk];
      a.y = aRow[kk + 1];

      const float2* wrow = Wp + ((k >> 1) + half) * WP_STRIDE + l15;
      float2 w;
      v2f b;

      w = wrow[0];
      b.x = w.x; b.y = w.y;
      acc0 = __builtin_amdgcn_wmma_f32_16x16x4_f32(false, a, false, b,
                                                   (short)0, acc0, false, false);
      w = wrow[16];
      b.x = w.x; b.y = w.y;
      acc1 = __builtin_amdgcn_wmma_f32_16x16x4_f32(false, a, false, b,
                                                   (short)0, acc1, false, false);
      w = wrow[32];
      b.x = w.x; b.y = w.y;
      acc2 = __builtin_amdgcn_wmma_f32_16x16x4_f32(false, a, false, b,
                                                   (short)0, acc2, false, false);
      w = wrow[48];
      b.x = w.x; b.y = w.y;
      acc3 = __builtin_amdgcn_wmma_f32_16x16x4_f32(false, a, false, b,
                                                   (short)0, acc3, false, false);
    }

    // Epilogue: C/D layout -> VGPR r holds M = r + 8*half, N = l15 (+tile).
#pragma unroll
    for (int r = 0; r < 8; ++r) {
      const int row = rowBase + r + 8 * half;
      float* o = out + (size_t)row * 64 + l15;
      o[0]  = acc0[r] + b0;
      o[16] = acc1[r] + b1;
      o[32] = acc2[r] + b2;
      o[48] = acc3[r] + b3;
    }
  }
}

// -------------------------------------------------------------------------
// Zero-fill (agg buffer must be zeroed every call for determinism).
// -------------------------------------------------------------------------
__global__ __launch_bounds__(256) void zero_f32(float* __restrict__ p,
                                                size_t n) {
  size_t i = (size_t)blockIdx.x * blockDim.x + threadIdx.x;
  if (i < n) p[i] = 0.0f;
}

// -------------------------------------------------------------------------
// Edge push: 32 threads per edge, each thread owns one float2 feature pair.
// agg[dst[e]][2f..2f+1] += h[src[e]][2f..2f+1]. Gathers are coalesced 256B
// rows; both gather and f32 atomics land in L2 (h/agg = 25.6 MB each, well
// inside the 192 MB L2).
// -------------------------------------------------------------------------
__global__ __launch_bounds__(256) void scatter_edges(
    const float* __restrict__ h, const int* __restrict__ esrc,
    const int* __restrict__ edst, float* __restrict__ agg, int nEdges) {
  const long long gid = (long long)blockIdx.x * blockDim.x + threadIdx.x;
  const int e  = (int)(gid >> 5);
  const int fp = (int)(gid & 31);  // feature-pair index
  if (e >= nEdges) return;
  const int s = esrc[e];
  const int d = edst[e];
  const float2 v = ((const float2*)h)[(size_t)s * 32 + fp];
  float* dst = agg + (size_t)d * 64 + 2 * fp;
  unsafeAtomicAdd(dst + 0, v.x);  // -> global_atomic_add_f32
  unsafeAtomicAdd(dst + 1, v.y);
}

// -------------------------------------------------------------------------
// Launch: data -> (GEMM1) -> h[ws] -> (scatter) -> agg(=d_out, zeroed)
//         -> (GEMM2, safely in-place) -> d_out
// -------------------------------------------------------------------------
extern "C" void kernel_launch(void* const* d_in, const int* in_sizes, int n_in,
                              void* d_out, int out_size, void* d_ws,
                              size_t ws_size, hipStream_t stream) {
  const float* data   = (const float*)d_in[0];
  const int*   esrc   = (const int*)d_in[1];
  const int*   edst   = (const int*)d_in[2];
  const float* W_pre  = (const float*)d_in[3];
  const float* b_pre  = (const float*)d_in[4];
  const float* W_post = (const float*)d_in[5];
  const float* b_post = (const float*)d_in[6];

  float* out = (float*)d_out;
  float* h   = (float*)d_ws;  // N_NODES * D_HID floats = 25.6 MB
  float* agg = out;           // reuse output buffer as accumulator

  // 1) zero accumulator (== d_out)
  {
    const size_t n = (size_t)N_NODES * D_HID;
    zero_f32<<<(unsigned)((n + 255) / 256), 256, 0, stream>>>(agg, n);
  }

  // 2) h = data @ W_pre + b_pre (K = 128); persistent blocks amortize the
  //    LDS staging of W (6250 tiles over 2000 waves).
  {
    const int blocks = 250;
    gemm_wmma_f32<D_IN><<<blocks, 256, 0, stream>>>(data, W_pre, b_pre, h,
                                                    N_NODES, blocks);
  }

  // 3) agg[dst] += h[src] over edges (32 threads / edge, float2 pairs)
  {
    const long long work   = (long long)N_EDGES * 32;
    const int       blocks = (int)((work + 255) / 256);  // 200000
    scatter_edges<<<blocks, 256, 0, stream>>>(h, esrc, edst, agg, N_EDGES);
  }

  // 4) out = agg @ W_post + b_post (K = 64, in-place on d_out: each wave
  //    fully reads its 16 rows into registers before storing them)
  {
    const int blocks = 250;
    gemm_wmma_f32<D_HID><<<blocks, 256, 0, stream>>>(agg, W_post, b_post, out,
                                                     N_NODES, blocks);
  }
}